// VectorQuantizer_45715631898663
// MI455X (gfx1250) — compile-verified
//
#include <hip/hip_runtime.h>
#include <stdint.h>

typedef __attribute__((ext_vector_type(2))) float v2f;
typedef __attribute__((ext_vector_type(8))) float v8f;

#define BT         8192              // B*T rows
#define DIM        64
#define NCODES     512
#define NBLOCKS    (NCODES / 16)     // 32 column tiles
#define KSTEPS     (DIM / 4)         // 16 WMMA K-steps of 4
#define ROW_STRIDE 68                // 64 + 4 pad floats; stride%64==4 -> conflict-free ds_load_b64
#define NORM_OFF   (NCODES * ROW_STRIDE)
#define LDS_FLOATS (NORM_OFF + NCODES)
#define RTILES     2                 // row tiles (16 rows each) per wave

#define USE_ASYNC_LDS 1              // flip to 0 if the async mnemonics fail to assemble

// 32 blocks x 256 threads. Each block: stage codebook -> LDS (async b128),
// precompute 0.5*||c||^2, then each wave runs 2 row-tiles x 512 codes of
// V_WMMA_F32_16X16X4_F32 argmax(x.c - 0.5||c||^2) == L2 argmin.
__global__ __launch_bounds__(256)
void vq_wmma_f32_kernel(const float* __restrict__ X,   // [BT, 64]
                        const float* __restrict__ C,   // [512, 64]
                        float* __restrict__ out_k,     // [BT]  (indices as float)
                        float* __restrict__ out_zq) {  // [BT, 64]
    __shared__ float ldsC[LDS_FLOATS];

    const int tid  = threadIdx.x;
    const int lane = tid & 31;
    const int nlo  = lane & 15;     // row (A) / col (B) / N (C) within tile
    const int half = lane >> 4;     // 0: K%4 in {0,1}, 1: K%4 in {2,3}
    const int koff = half * 2;

    // ---- Stage codebook (8192 x 16B chunks) into padded LDS rows ----
    {
        const uint32_t lds_base = (uint32_t)(uintptr_t)(&ldsC[0]);
        const uint64_t gbase    = (uint64_t)(uintptr_t)C;
#pragma unroll 4
        for (int i = 0; i < 32; ++i) {
            const int      cid  = i * 256 + tid;          // 16B chunk id
            const uint32_t crow = (uint32_t)(cid >> 4);
            const uint32_t cch  = (uint32_t)(cid & 15);
            const uint32_t goff = (uint32_t)cid * 16u;
            const uint32_t loff = lds_base + crow * (ROW_STRIDE * 4u) + cch * 16u;
#if USE_ASYNC_LDS
            asm volatile("global_load_async_to_lds_b128 %0, %1, %2"
                         :: "v"(loff), "v"(goff), "s"(gbase) : "memory");
#else
            const float4 t = *reinterpret_cast<const float4*>(C + cid * 4);
            *reinterpret_cast<float4*>(&ldsC[crow * ROW_STRIDE + cch * 4]) = t;
#endif
        }
#if USE_ASYNC_LDS
        asm volatile("s_wait_asynccnt 0" ::: "memory");
#endif
    }
    __syncthreads();

    // ---- Precompute 0.5*||c||^2 per code (once per block) ----
    for (int c = tid; c < NCODES; c += 256) {
        float s = 0.0f;
#pragma unroll
        for (int k = 0; k < DIM; k += 2) {
            v2f t = *reinterpret_cast<const v2f*>(&ldsC[c * ROW_STRIDE + k]);
            s += t.x * t.x + t.y * t.y;
        }
        ldsC[NORM_OFF + c] = 0.5f * s;
    }
    __syncthreads();

    // ---- Load A tiles: 2 x 16 rows, this lane holds K = 4*kk + koff + {0,1} ----
    const int wave    = (blockIdx.x * 256 + tid) >> 5;   // 0..255
    const int rowbase = wave * (RTILES * 16);
    v2f a[RTILES][KSTEPS];
#pragma unroll
    for (int r = 0; r < RTILES; ++r)
#pragma unroll
        for (int kk = 0; kk < KSTEPS; ++kk)
            a[r][kk] = *reinterpret_cast<const v2f*>(
                X + (rowbase + r * 16 + nlo) * DIM + kk * 4 + koff);

    float best_s[RTILES][8];
    int   best_i[RTILES][8];
#pragma unroll
    for (int r = 0; r < RTILES; ++r)
#pragma unroll
        for (int v = 0; v < 8; ++v) { best_s[r][v] = -3.402823466e38f; best_i[r][v] = 0; }

    for (int nb = 0; nb < NBLOCKS; ++nb) {
        const int col = nb * 16 + nlo;
        // B tile from LDS: conflict-free (row stride 68 -> 4 banks/lane, 64 banks covered)
        v2f b[KSTEPS];
#pragma unroll
        for (int kk = 0; kk < KSTEPS; ++kk)
            b[kk] = *reinterpret_cast<const v2f*>(&ldsC[col * ROW_STRIDE + kk * 4 + koff]);
        const float negcn = -ldsC[NORM_OFF + col];

#pragma unroll
        for (int r = 0; r < RTILES; ++r) {
            v8f acc;
#pragma unroll
            for (int v = 0; v < 8; ++v) acc[v] = negcn;
#pragma unroll
            for (int kk = 0; kk < KSTEPS; ++kk) {
                acc = __builtin_amdgcn_wmma_f32_16x16x4_f32(
                    false, a[r][kk], false, b[kk], (short)0, acc, false, false);
            }
#pragma unroll
            for (int v = 0; v < 8; ++v) {
                const float s = acc[v];
                if (s > best_s[r][v]) { best_s[r][v] = s; best_i[r][v] = col; }
            }
        }
    }

    // ---- Reduce across the 16 lanes of each half-wave (masks <16 stay in-half) ----
#pragma unroll
    for (int mask = 1; mask < 16; mask <<= 1) {
#pragma unroll
        for (int r = 0; r < RTILES; ++r)
#pragma unroll
            for (int v = 0; v < 8; ++v) {
                const float os = __shfl_xor(best_s[r][v], mask, 32);
                const int   oi = __shfl_xor(best_i[r][v], mask, 32);
                if (os > best_s[r][v] || (os == best_s[r][v] && oi < best_i[r][v])) {
                    best_s[r][v] = os; best_i[r][v] = oi;
                }
            }
    }

    // ---- Emit: 16 lanes/half copy the winning code row (float4 each) from LDS ----
#pragma unroll
    for (int r = 0; r < RTILES; ++r)
#pragma unroll
        for (int v = 0; v < 8; ++v) {
            const int m    = v + half * 8;
            const int grow = rowbase + r * 16 + m;
            const int idx  = best_i[r][v];
            if (nlo == 0) out_k[grow] = (float)idx;
            const float4 zq = *(reinterpret_cast<const float4*>(&ldsC[idx * ROW_STRIDE]) + nlo);
            *(reinterpret_cast<float4*>(out_zq + grow * DIM) + nlo) = zq;
        }
}

extern "C" void kernel_launch(void* const* d_in, const int* in_sizes, int n_in,
                              void* d_out, int out_size, void* d_ws, size_t ws_size,
                              hipStream_t stream) {
    const float* X = (const float*)d_in[0];   // [4,2048,64] fp32
    const float* C = (const float*)d_in[1];   // [512,64]    fp32
    float* out    = (float*)d_out;            // [8192] k ++ [8192*64] z_q
    float* out_k  = out;
    float* out_zq = out + BT;

    // 512 row-tiles; 8 waves/block x 2 tiles/wave -> 32 blocks x 256 threads
    const int blocks = BT / (256 * RTILES / 32 * 16); // = 32
    vq_wmma_f32_kernel<<<blocks, 256, 0, stream>>>(X, C, out_k, out_zq);
}